// AttentionBlock_55121610276881
// MI455X (gfx1250) — compile-verified
//
#include <hip/hip_runtime.h>
#include <hip/hip_bf16.h>

typedef __attribute__((ext_vector_type(2))) float v2f;
typedef __attribute__((ext_vector_type(8))) float v8f;

#define Bx 2
#define Hx 128
#define Wx 128
#define Cx 4
#define Nx (Hx * Wx)          // 16384
#define DHx (Hx / 2)
#define DWx (Wx / 2)
#define Mx (DHx * DWx)        // 4096
#define NTILES (Nx / 16)      // 1024
#define WAVES_PER_BLOCK 8
#define BLOCKS_PER_BATCH (NTILES / WAVES_PER_BLOCK)  // 128
#define MCHUNK 1024
#define MCHUNK_TILES (MCHUNK / 16)                   // 64
#define NCHUNKS (Mx / MCHUNK)                        // 4

// ---- async global->LDS copy (CDNA5 ASYNCcnt path) ----
// GVS form: LDS address from vdst, global address from saddr + vaddr; both add the immediate offset
// The immediate offset advances LDS and global addresses in lockstep, so the
// address VGPRs stay constant while ops are in flight (no XCNT hazard).
#define ASYNC_CP_B128(lds_addr, glob_off, base, IMM)                        \
  asm volatile("global_load_async_to_lds_b128 %0, %1, %2 offset:" #IMM      \
               :                                                            \
               : "v"(lds_addr), "v"(glob_off), "s"(base)                    \
               : "memory")

__device__ __forceinline__ void wait_async_lds() {
  asm volatile("s_wait_asynccnt 0x0" ::: "memory");
}
__device__ __forceinline__ unsigned lds_off32(const void* p) {
  // generic pointer to LDS = {shared_aperture_hi32, lds_offset_lo32}
  return (unsigned)(uintptr_t)p;
}

// ---- float <-> order-preserving unsigned encoding (for atomic max) ----
__device__ __forceinline__ unsigned enc_f32(float x) {
  unsigned b = __float_as_uint(x);
  return (b & 0x80000000u) ? ~b : (b | 0x80000000u);
}
__device__ __forceinline__ float dec_f32(unsigned e) {
  unsigned b = (e & 0x80000000u) ? (e & 0x7FFFFFFFu) : ~e;
  return __uint_as_float(b);
}

// ---------------- init scalars ----------------
__global__ void k_init(unsigned* __restrict__ maxenc, float* __restrict__ Z) {
  int t = threadIdx.x;
  if (t < Bx) { maxenc[t] = 0u; Z[t] = 0.0f; }
}

// ---------------- f = x @ w1 + b1  (full res) ----------------
__global__ __launch_bounds__(256) void k_proj_f(const float* __restrict__ x,
                                                const float* __restrict__ w1,
                                                const float* __restrict__ b1,
                                                float* __restrict__ f) {
  int gid = blockIdx.x * 256 + threadIdx.x;  // 0 .. B*N-1
  float4 xv = *(const float4*)(x + (size_t)gid * 4);
  float* fo = f + (size_t)gid * 4;
#pragma unroll
  for (int d = 0; d < 4; ++d) {
    float v = b1[d];
    v = fmaf(xv.x, w1[0 * 4 + d], v);
    v = fmaf(xv.y, w1[1 * 4 + d], v);
    v = fmaf(xv.z, w1[2 * 4 + d], v);
    v = fmaf(xv.w, w1[3 * 4 + d], v);
    fo[d] = v;
  }
}

// ---------------- g = maxpool(x @ w2 + b2), phi = maxpool(x @ w3 + b3) ----------------
__global__ __launch_bounds__(256) void k_proj_gphi(const float* __restrict__ x,
                                                   const float* __restrict__ w2,
                                                   const float* __restrict__ b2,
                                                   const float* __restrict__ w3,
                                                   const float* __restrict__ b3,
                                                   float* __restrict__ g,
                                                   float* __restrict__ phi) {
  int gid = blockIdx.x * 256 + threadIdx.x;  // 0 .. B*M-1
  int b = gid / Mx;
  int pm = gid % Mx;
  int ph = pm / DWx, pw = pm % DWx;
  float gm[4], pmv[4];
#pragma unroll
  for (int d = 0; d < 4; ++d) { gm[d] = -3.402823466e38f; pmv[d] = -3.402823466e38f; }
#pragma unroll
  for (int di = 0; di < 2; ++di) {
#pragma unroll
    for (int dj = 0; dj < 2; ++dj) {
      int h = ph * 2 + di, w = pw * 2 + dj;
      float4 xv = *(const float4*)(x + ((size_t)((b * Hx + h) * Wx + w)) * 4);
#pragma unroll
      for (int d = 0; d < 4; ++d) {
        float gv = b2[d];
        gv = fmaf(xv.x, w2[0 * 4 + d], gv);
        gv = fmaf(xv.y, w2[1 * 4 + d], gv);
        gv = fmaf(xv.z, w2[2 * 4 + d], gv);
        gv = fmaf(xv.w, w2[3 * 4 + d], gv);
        gm[d] = fmaxf(gm[d], gv);
        float pv = b3[d];
        pv = fmaf(xv.x, w3[0 * 4 + d], pv);
        pv = fmaf(xv.y, w3[1 * 4 + d], pv);
        pv = fmaf(xv.z, w3[2 * 4 + d], pv);
        pv = fmaf(xv.w, w3[3 * 4 + d], pv);
        pmv[d] = fmaxf(pmv[d], pv);
      }
    }
  }
#pragma unroll
  for (int d = 0; d < 4; ++d) {
    g[(size_t)gid * 4 + d] = gm[d];
    phi[(size_t)gid * 4 + d] = pmv[d];
  }
}

// ---------------- pass 1: global max of f @ g^T via WMMA ----------------
__global__ __launch_bounds__(256) void k_attn_max(const float* __restrict__ f,
                                                  const float* __restrict__ g,
                                                  unsigned* __restrict__ maxenc) {
  __shared__ float gl[MCHUNK * Cx];  // 16 KB
  int b = blockIdx.x / BLOCKS_PER_BATCH;
  int nblk = blockIdx.x % BLOCKS_PER_BATCH;
  int tid = threadIdx.x;
  int wave = tid >> 5, lane = tid & 31;
  int row = lane & 15;
  int k0 = (lane >> 4) << 1;  // lanes 0-15: K=0,1 ; lanes 16-31: K=2,3 (ISA A/B layout)
  int n0 = (nblk * WAVES_PER_BLOCK + wave) * 16;

  const float* frow = f + ((size_t)(b * Nx + n0 + row) * Cx + k0);
  v2f A;
  A.x = frow[0];
  A.y = frow[1];

  const float* gb = g + (size_t)b * Mx * Cx;
  unsigned voff = (unsigned)tid * 16u;        // per-lane byte offset, constant
  unsigned gld = lds_off32(gl) + voff;        // per-lane LDS byte addr, constant
  float vmax = -3.402823466e38f;

  for (int ch = 0; ch < NCHUNKS; ++ch) {
    __syncthreads();  // previous chunk fully consumed before overwrite
    const float* gsrc = gb + (size_t)ch * MCHUNK * Cx;
    // 1024 rows * 16B = 16 KB per chunk; 256 threads x 16B = 4 KB per issue
    ASYNC_CP_B128(gld, voff, gsrc, 0);
    ASYNC_CP_B128(gld, voff, gsrc, 4096);
    ASYNC_CP_B128(gld, voff, gsrc, 8192);
    ASYNC_CP_B128(gld, voff, gsrc, 12288);
    wait_async_lds();
    __syncthreads();
#pragma unroll 4
    for (int jt = 0; jt < MCHUNK_TILES; ++jt) {
      int mi = jt * 16 + row;
      v2f Bv;
      Bv.x = gl[mi * 4 + k0];
      Bv.y = gl[mi * 4 + k0 + 1];
      v8f c = {};
      c = __builtin_amdgcn_wmma_f32_16x16x4_f32(false, A, false, Bv, (short)0, c,
                                                false, false);
#pragma unroll
      for (int v = 0; v < 8; ++v) vmax = fmaxf(vmax, c[v]);
    }
  }
#pragma unroll
  for (int off = 1; off < 32; off <<= 1)
    vmax = fmaxf(vmax, __shfl_xor(vmax, off, 32));
  if (lane == 0) atomicMax(&maxenc[b], enc_f32(vmax));
}

// ---------------- pass 2: S[n,c] = sum_m exp(a-max)*phi[m,c], Z = sum exp ----------------
__global__ __launch_bounds__(256) void k_attn_sv(const float* __restrict__ f,
                                                 const float* __restrict__ g,
                                                 const float* __restrict__ phi,
                                                 const unsigned* __restrict__ maxenc,
                                                 float* __restrict__ S,
                                                 float* __restrict__ Z) {
  __shared__ float gl[MCHUNK * Cx];  // 16 KB
  __shared__ float pl[MCHUNK * Cx];  // 16 KB
  int b = blockIdx.x / BLOCKS_PER_BATCH;
  int nblk = blockIdx.x % BLOCKS_PER_BATCH;
  int tid = threadIdx.x;
  int wave = tid >> 5, lane = tid & 31;
  int row = lane & 15;
  int k0 = (lane >> 4) << 1;
  int n0 = (nblk * WAVES_PER_BLOCK + wave) * 16;

  const float* frow = f + ((size_t)(b * Nx + n0 + row) * Cx + k0);
  v2f A;
  A.x = frow[0];
  A.y = frow[1];

  float gmax = dec_f32(maxenc[b]);
  const float* gb = g + (size_t)b * Mx * Cx;
  const float* pb = phi + (size_t)b * Mx * Cx;
  unsigned voff = (unsigned)tid * 16u;
  unsigned gld = lds_off32(gl) + voff;
  unsigned pld = lds_off32(pl) + voff;

  float acc[32];
#pragma unroll
  for (int i = 0; i < 32; ++i) acc[i] = 0.0f;
  float zsum = 0.0f;

  for (int ch = 0; ch < NCHUNKS; ++ch) {
    __syncthreads();  // previous chunk fully consumed before overwrite
    const float* gsrc = gb + (size_t)ch * MCHUNK * Cx;
    const float* psrc = pb + (size_t)ch * MCHUNK * Cx;
    ASYNC_CP_B128(gld, voff, gsrc, 0);
    ASYNC_CP_B128(gld, voff, gsrc, 4096);
    ASYNC_CP_B128(gld, voff, gsrc, 8192);
    ASYNC_CP_B128(gld, voff, gsrc, 12288);
    ASYNC_CP_B128(pld, voff, psrc, 0);
    ASYNC_CP_B128(pld, voff, psrc, 4096);
    ASYNC_CP_B128(pld, voff, psrc, 8192);
    ASYNC_CP_B128(pld, voff, psrc, 12288);
    wait_async_lds();
    __syncthreads();
#pragma unroll 2
    for (int jt = 0; jt < MCHUNK_TILES; ++jt) {
      int mi = jt * 16 + row;
      v2f Bv;
      Bv.x = gl[mi * 4 + k0];
      Bv.y = gl[mi * 4 + k0 + 1];
      v8f c = {};
      c = __builtin_amdgcn_wmma_f32_16x16x4_f32(false, A, false, Bv, (short)0, c,
                                                false, false);
      // lane (either half) owns score column m = jt*16 + (lane&15); rows v (+8 for hi half)
      float4 ph = *(const float4*)(pl + mi * 4);
#pragma unroll
      for (int v = 0; v < 8; ++v) {
        float e = __expf(c[v] - gmax);
        zsum += e;
        acc[4 * v + 0] = fmaf(e, ph.x, acc[4 * v + 0]);
        acc[4 * v + 1] = fmaf(e, ph.y, acc[4 * v + 1]);
        acc[4 * v + 2] = fmaf(e, ph.z, acc[4 * v + 2]);
        acc[4 * v + 3] = fmaf(e, ph.w, acc[4 * v + 3]);
      }
    }
  }
  // reduce over the 16 m-columns within each half-wave (xor masks stay inside halves)
#pragma unroll
  for (int off = 1; off < 16; off <<= 1) {
#pragma unroll
    for (int i = 0; i < 32; ++i) acc[i] += __shfl_xor(acc[i], off, 32);
  }
#pragma unroll
  for (int off = 1; off < 32; off <<= 1) zsum += __shfl_xor(zsum, off, 32);
  if (lane == 0) atomicAdd(&Z[b], zsum);
  if ((lane & 15) == 0) {
    int rbase = n0 + (lane >> 4) * 8;  // lane0 -> rows 0..7, lane16 -> rows 8..15
    float* dst = S + (size_t)(b * Nx + rbase) * Cx;
#pragma unroll
    for (int v = 0; v < 8; ++v) {
#pragma unroll
      for (int c4 = 0; c4 < 4; ++c4) dst[v * 4 + c4] = acc[4 * v + c4];
    }
  }
}

// ---------------- y = x + sigma * ((S/Z) @ wl + bl) ----------------
__global__ __launch_bounds__(256) void k_apply(const float* __restrict__ x,
                                               const float* __restrict__ S,
                                               const float* __restrict__ Z,
                                               const float* __restrict__ wl,
                                               const float* __restrict__ bl,
                                               const float* __restrict__ sigma,
                                               float* __restrict__ y) {
  int gid = blockIdx.x * 256 + threadIdx.x;  // 0 .. B*N-1
  int b = gid / Nx;
  float invZ = 1.0f / Z[b];
  float4 s = *(const float4*)(S + (size_t)gid * 4);
  float a0 = s.x * invZ, a1 = s.y * invZ, a2 = s.z * invZ, a3 = s.w * invZ;
  float4 xv = *(const float4*)(x + (size_t)gid * 4);
  float sg = sigma[0];
  float* yo = y + (size_t)gid * 4;
  float xs[4] = {xv.x, xv.y, xv.z, xv.w};
#pragma unroll
  for (int d = 0; d < 4; ++d) {
    float v = bl[d];
    v = fmaf(a0, wl[0 * 4 + d], v);
    v = fmaf(a1, wl[1 * 4 + d], v);
    v = fmaf(a2, wl[2 * 4 + d], v);
    v = fmaf(a3, wl[3 * 4 + d], v);
    yo[d] = fmaf(sg, v, xs[d]);
  }
}

// ---------------- out = conv6x6 SAME (pad 2 top/left, 3 bottom/right) + bo ----------------
__global__ __launch_bounds__(256) void k_conv(const float* __restrict__ y,
                                              const float* __restrict__ wo,
                                              const float* __restrict__ bo,
                                              float* __restrict__ out) {
  __shared__ float wsm[6 * 6 * 4 * 4];  // 2.25 KB
  int tid = threadIdx.x;
  for (int i = tid; i < 576; i += 256) wsm[i] = wo[i];
  __syncthreads();
  int gid = blockIdx.x * 256 + tid;  // 0 .. B*H*W-1
  int b = gid / (Hx * Wx);
  int p = gid % (Hx * Wx);
  int h = p / Wx, w = p % Wx;
  float acc[4] = {bo[0], bo[1], bo[2], bo[3]};
#pragma unroll
  for (int kh = 0; kh < 6; ++kh) {
    int hy = h + kh - 2;
    if ((unsigned)hy >= (unsigned)Hx) continue;
#pragma unroll
    for (int kw = 0; kw < 6; ++kw) {
      int wy = w + kw - 2;
      if ((unsigned)wy >= (unsigned)Wx) continue;
      float4 yv = *(const float4*)(y + ((size_t)((b * Hx + hy) * Wx + wy)) * 4);
      const float* wp = wsm + (kh * 6 + kw) * 16;  // [ci][co]
#pragma unroll
      for (int co = 0; co < 4; ++co) {
        float v = acc[co];
        v = fmaf(yv.x, wp[0 * 4 + co], v);
        v = fmaf(yv.y, wp[1 * 4 + co], v);
        v = fmaf(yv.z, wp[2 * 4 + co], v);
        v = fmaf(yv.w, wp[3 * 4 + co], v);
        acc[co] = v;
      }
    }
  }
  float* op = out + (size_t)gid * 4;
#pragma unroll
  for (int co = 0; co < 4; ++co) op[co] = acc[co];
}

extern "C" void kernel_launch(void* const* d_in, const int* in_sizes, int n_in,
                              void* d_out, int out_size, void* d_ws, size_t ws_size,
                              hipStream_t stream) {
  const float* x = (const float*)d_in[0];
  const float* w1 = (const float*)d_in[1];
  const float* b1 = (const float*)d_in[2];
  const float* w2 = (const float*)d_in[3];
  const float* b2 = (const float*)d_in[4];
  const float* w3 = (const float*)d_in[5];
  const float* b3 = (const float*)d_in[6];
  const float* wl = (const float*)d_in[7];
  const float* bl = (const float*)d_in[8];
  const float* wo = (const float*)d_in[9];
  const float* bo = (const float*)d_in[10];
  const float* sigma = (const float*)d_in[11];
  float* out = (float*)d_out;

  char* ws = (char*)d_ws;
  float* f = (float*)(ws + 0);                 // B*N*C = 512 KB
  float* g = (float*)(ws + 524288);            // B*M*C = 128 KB
  float* phi = (float*)(ws + 655360);          // 128 KB
  float* S = (float*)(ws + 786432);            // 512 KB
  float* y = (float*)(ws + 1310720);           // 512 KB
  unsigned* maxenc = (unsigned*)(ws + 1835008);
  float* Z = (float*)(ws + 1835008 + 64);

  k_init<<<1, 32, 0, stream>>>(maxenc, Z);
  k_proj_f<<<(Bx * Nx) / 256, 256, 0, stream>>>(x, w1, b1, f);
  k_proj_gphi<<<(Bx * Mx) / 256, 256, 0, stream>>>(x, w2, b2, w3, b3, g, phi);
  k_attn_max<<<Bx * BLOCKS_PER_BATCH, 256, 0, stream>>>(f, g, maxenc);
  k_attn_sv<<<Bx * BLOCKS_PER_BATCH, 256, 0, stream>>>(f, g, phi, maxenc, S, Z);
  k_apply<<<(Bx * Nx) / 256, 256, 0, stream>>>(x, S, Z, wl, bl, sigma, y);
  k_conv<<<(Bx * Hx * Wx) / 256, 256, 0, stream>>>(y, wo, bo, out);
}